// NNDModule_42210938585527
// MI455X (gfx1250) — compile-verified
//
#include <hip/hip_runtime.h>
#include <math.h>

// CDNA5 / gfx1250 Chamfer distance via V_WMMA_F32_16X16X4_F32 (wave32).
//
// out[q] = max(min_p ||P_p - Q_q||^2, 0) per batch slice; launched twice with
// (P,Q) swapped to produce dist1 and dist2.
//
// WMMA encoding of the distance expansion (K=4):
//   A[i] = (-2p0, -2p1, -2p2, ||p||^2),  B[j] = (q0, q1, q2, 1),  C[.,j] = ||q_j||^2
//   => D[i][j] = ||p_i||^2 - 2 p_i.q_j + ||q_j||^2   (full squared distance)
//
// A-operands are pre-transformed into WMMA register layout in LDS so the inner
// loop is: ds_load_b64 -> v_wmma -> v_min3 tree. Clamp to 0 is hoisted out.

typedef float v2f __attribute__((ext_vector_type(2)));
typedef float v8f __attribute__((ext_vector_type(8)));

#define WAVES_PER_BLOCK 8   // 256 threads, 8 wave32s
#define TILE 16

__global__ void chamfer_min_wmma(const float* __restrict__ P,
                                 const float* __restrict__ Q,
                                 float* __restrict__ out,
                                 int NP, int NQ) {
    // sA: one 64-float record per P-tile, already in A-matrix lane layout:
    //   record[t][lane*2 + {0,1}] = this lane's (K-pair) for tile t.
    extern __shared__ float sA[];             // (NP/16) * 64 floats = NP*16 bytes

    // Problem sizes are multiples of the full block tile; lets the compiler
    // drop the scalar remainder loops.
    __builtin_assume(NP >= 64 && (NP & 63) == 0);
    __builtin_assume(NQ >= 128 && (NQ & 127) == 0);

    const int b    = blockIdx.y;
    const int tid  = threadIdx.x;
    const int lane = tid & 31;
    const int wid  = tid >> 5;
    const int hi   = lane >> 4;               // lane half selector
    const int l16  = lane & 15;

    // ---- Stage + transform this batch's P slab into WMMA A layout ----
    const float* Pb = P + (size_t)b * NP * 3;
    for (int i = tid; i < NP; i += blockDim.x) {
        float p0 = Pb[3 * i + 0];
        float p1 = Pb[3 * i + 1];
        float p2 = Pb[3 * i + 2];
        float n  = fmaf(p0, p0, fmaf(p1, p1, p2 * p2));
        float* slot = &sA[(i >> 4) * 64];
        int r = i & 15;
        slot[r * 2 + 0]        = -2.0f * p0;  // lane r      : (K0, K1)
        slot[r * 2 + 1]        = -2.0f * p1;
        slot[(r + 16) * 2 + 0] = -2.0f * p2;  // lane r + 16 : (K2, K3)
        slot[(r + 16) * 2 + 1] = n;
    }
    __syncthreads();

    // ---- Per-wave Q tile (wave-uniform guard keeps EXEC all-ones for WMMA) ----
    const int qtile = blockIdx.x * WAVES_PER_BLOCK + wid;
    if (qtile * TILE >= NQ) return;

    const int j = qtile * TILE + l16;         // this lane's Q column
    const float* Qb = Q + (size_t)b * NQ * 3;

    // B operand: lanes 0-15 -> (q0,q1) = (K0,K1); lanes 16-31 -> (q2,1) = (K2,K3)
    v2f bmat;
    float qpart;
    if (!hi) {
        float q0 = Qb[3 * j + 0];
        float q1 = Qb[3 * j + 1];
        bmat.x = q0; bmat.y = q1;
        qpart  = fmaf(q0, q0, q1 * q1);
    } else {
        float q2 = Qb[3 * j + 2];
        bmat.x = q2; bmat.y = 1.0f;
        qpart  = q2 * q2;
    }
    const float q2n = qpart + __shfl_xor(qpart, 16, 32);

    // C operand: broadcast ||q_j||^2 into all 8 row-VGPRs (col j = lane%16)
    v8f cvec;
#pragma unroll
    for (int r = 0; r < 8; ++r) cvec[r] = q2n;

    float best = __builtin_inff();
    const int ntiles = NP >> 4;               // multiple of 4 (NP % 64 == 0)
    const v2f* sAv = (const v2f*)sA;          // 32 v2f records per tile

#pragma unroll 4
    for (int t = 0; t < ntiles; ++t) {
        v2f amat = sAv[t * 32 + lane];        // single ds_load_b64, conflict-free
        // D = A x B + C : full 16x16 tile of squared distances
        v8f d = __builtin_amdgcn_wmma_f32_16x16x4_f32(
            /*neg_a=*/false, amat, /*neg_b=*/false, bmat,
            /*c_mod=*/(short)0, cvec, /*reuse_a=*/false, /*reuse_b=*/false);

        // min3-shaped tree: 4x v_min3_num_f32, depth-1 chain into `best`
        float t0 = fminf(fminf(d[0], d[1]), d[2]);
        float t1 = fminf(fminf(d[3], d[4]), d[5]);
        float t2 = fminf(fminf(t0, t1), d[6]);
        best     = fminf(fminf(best, t2), d[7]);
    }

    // merge the two lane halves (rows 0-7 vs 8-15), then clamp once
    best = fminf(best, __shfl_xor(best, 16, 32));
    best = fmaxf(best, 0.0f);

    if (!hi) {
        out[(size_t)b * NQ + qtile * TILE + l16] = best;
    }
}

extern "C" void kernel_launch(void* const* d_in, const int* in_sizes, int n_in,
                              void* d_out, int out_size, void* d_ws, size_t ws_size,
                              hipStream_t stream) {
    const float* x = (const float*)d_in[0];   // [B, N, 3]
    const float* y = (const float*)d_in[1];   // [B, M, 3]
    float* out = (float*)d_out;               // dist1 [B,N] ++ dist2 [B,M]

    const int B = 8, D = 3;
    const int N = in_sizes[0] / (B * D);
    const int M = in_sizes[1] / (B * D);

    const int threads = WAVES_PER_BLOCK * 32;

    // Pass 1: dist1[n] = min_m ||x_n - y_m||^2   (P = y, Q = x)
    {
        dim3 grid((N + WAVES_PER_BLOCK * TILE - 1) / (WAVES_PER_BLOCK * TILE), B);
        size_t shmem = (size_t)M * 16;        // (M/16) tiles * 64 floats
        chamfer_min_wmma<<<grid, threads, shmem, stream>>>(y, x, out, M, N);
    }
    // Pass 2: dist2[m] = min_n ||y_m - x_n||^2   (P = x, Q = y)
    {
        dim3 grid((M + WAVES_PER_BLOCK * TILE - 1) / (WAVES_PER_BLOCK * TILE), B);
        size_t shmem = (size_t)N * 16;
        chamfer_min_wmma<<<grid, threads, shmem, stream>>>(x, y, out + (size_t)B * N, N, M);
    }
}